// MTS_49460843381229
// MI455X (gfx1250) — compile-verified
//
#include <hip/hip_runtime.h>
#include <math.h>

typedef __attribute__((ext_vector_type(2))) float v2f;
typedef __attribute__((ext_vector_type(8))) float v8f;

__device__ __forceinline__ v8f vzero8() {
    v8f v; for (int i = 0; i < 8; ++i) v[i] = 0.0f; return v;
}

// D = A(16x4) * B(4x16) + C, fp32 WMMA (CDNA5)
__device__ __forceinline__ v8f wmma4(v2f a, v2f b, v8f c) {
    return __builtin_amdgcn_wmma_f32_16x16x4_f32(false, a, false, b, (short)0, c, false, false);
}

__device__ __forceinline__ float gelu_exact(float v) {
    return 0.5f * v * (1.0f + erff(v * 0.70710678118654752f));
}

#define PS 24576  // per-batch position stride (12*2048)

// ---------------------------------------------------------------------------
// A = softmax(relu(E @ E^T)) per row; E is (2048,10)
__global__ void adj_k(const float* __restrict__ emb, float* __restrict__ Adj) {
    __shared__ float red[256];
    __shared__ float en[10];
    int n = blockIdx.x, tid = threadIdx.x;
    if (tid < 10) en[tid] = emb[n * 10 + tid];
    __syncthreads();
    float* row = Adj + (long)n * 2048;
    float lmax = 0.0f;
    for (int m = tid; m < 2048; m += 256) {
        float d = 0.0f;
        for (int dd = 0; dd < 10; ++dd) d += en[dd] * emb[m * 10 + dd];
        d = d > 0.0f ? d : 0.0f;
        row[m] = d;
        lmax = fmaxf(lmax, d);
    }
    red[tid] = lmax; __syncthreads();
    for (int s = 128; s > 0; s >>= 1) {
        if (tid < s) red[tid] = fmaxf(red[tid], red[tid + s]);
        __syncthreads();
    }
    float rmax = red[0];
    __syncthreads();
    float lsum = 0.0f;
    for (int m = tid; m < 2048; m += 256) {
        float e = expf(row[m] - rmax);
        row[m] = e; lsum += e;
    }
    red[tid] = lsum; __syncthreads();
    for (int s = 128; s > 0; s >>= 1) {
        if (tid < s) red[tid] += red[tid + s];
        __syncthreads();
    }
    float inv = 1.0f / red[0];
    for (int m = tid; m < 2048; m += 256) row[m] *= inv;
}

// X0[m*32+b] = x[b,11,m,0]
__global__ void pack_x0_k(const float* __restrict__ x, float* __restrict__ X0) {
    int i = blockIdx.x * 256 + threadIdx.x;   // b*2048+m
    int b = i >> 11, m = i & 2047;
    X0[m * 32 + b] = x[(((long)b * 12 + 11) * 2048 + m) * 3];
}

// Generic fp32 WMMA GEMM: C(MxN) = A(MxK) @ B(KxN), row-major, K%16==0, M,N%16==0.
// blockDim=128 (4 waves), each wave owns one 16x16 C tile; LDS-staged K chunks of 16.
__global__ void gemm_k(const float* __restrict__ A, const float* __restrict__ B,
                       float* __restrict__ C, int K, int N, int nTiles) {
    __shared__ float As[4][256];
    __shared__ float Bs[4][256];
    int tid = threadIdx.x, wv = tid >> 5, ln = tid & 31;
    int l16 = ln & 15, half = ln >> 4, klo = 2 * half;
    int tile = blockIdx.x * 4 + wv;
    int mt = tile / nTiles, nt = tile % nTiles;
    long m0 = (long)mt * 16, n0 = (long)nt * 16;
    v8f acc = vzero8();
    for (int kc = 0; kc < K; kc += 16) {
        const float4* pa = (const float4*)&A[(m0 + l16) * (long)K + kc + half * 8];
        const float4* pb = (const float4*)&B[(long)(kc + l16) * N + n0 + half * 8];
        float4 a0 = pa[0], a1 = pa[1];
        float4 b0 = pb[0], b1 = pb[1];
        ((float4*)&As[wv][l16 * 16 + half * 8])[0] = a0;
        ((float4*)&As[wv][l16 * 16 + half * 8])[1] = a1;
        ((float4*)&Bs[wv][l16 * 16 + half * 8])[0] = b0;
        ((float4*)&Bs[wv][l16 * 16 + half * 8])[1] = b1;
        __syncthreads();
        for (int kk = 0; kk < 16; kk += 4) {
            v2f a, b;
            a.x = As[wv][l16 * 16 + kk + klo];
            a.y = As[wv][l16 * 16 + kk + klo + 1];
            b.x = Bs[wv][(kk + klo) * 16 + l16];
            b.y = Bs[wv][(kk + klo + 1) * 16 + l16];
            acc = wmma4(a, b, acc);
        }
        __syncthreads();
    }
    for (int r = 0; r < 8; ++r) {
        int m = r + 8 * half;
        C[(m0 + m) * (long)N + n0 + l16] = acc[r];
    }
}

// GCN layer 0: Z[n][b*32+c] = X0[n,b]*W0a[n,c] + Y0[n,b]*W0b[n,c] + bias0[n,c]
__global__ void gcn0_k(const float* __restrict__ emb, const float* __restrict__ w0,
                       const float* __restrict__ b0, const float* __restrict__ X0,
                       const float* __restrict__ Y0, float* __restrict__ Z) {
    int n = blockIdx.x, c = threadIdx.x;     // block of 32 threads
    float wa = 0.0f, wb = 0.0f, bi = 0.0f;
    for (int d = 0; d < 10; ++d) {
        float e = emb[n * 10 + d];
        wa += e * w0[(d * 2 + 0) * 32 + c];
        wb += e * w0[(d * 2 + 1) * 32 + c];
        bi += e * b0[d * 32 + c];
    }
    for (int b = 0; b < 32; ++b) {
        float xv = X0[n * 32 + b], yv = Y0[n * 32 + b];
        Z[(long)n * 1024 + b * 32 + c] = xv * wa + yv * wb + bi;
    }
}

// GCN layer 1 -> h channels 0..31 (t=11 slice)
__global__ void gcn1_k(const float* __restrict__ emb, const float* __restrict__ w1,
                       const float* __restrict__ b1, const float* __restrict__ Z,
                       const float* __restrict__ Y1, float* __restrict__ h) {
    __shared__ float Wn[2048];
    __shared__ float Bn[32];
    int n = blockIdx.x, tid = threadIdx.x;
    float e[10];
    for (int d = 0; d < 10; ++d) e[d] = emb[n * 10 + d];
    for (int i = tid; i < 2048; i += 256) {
        float s = 0.0f;
        for (int d = 0; d < 10; ++d) s += e[d] * w1[d * 2048 + i];  // (10,2,32,32)
        Wn[i] = s;
    }
    if (tid < 32) {
        float s = 0.0f;
        for (int d = 0; d < 10; ++d) s += e[d] * b1[d * 32 + tid];
        Bn[tid] = s;
    }
    __syncthreads();
    for (int i = tid; i < 1024; i += 256) {
        int b = i >> 5, o = i & 31;
        float s = Bn[o];
        for (int c = 0; c < 32; ++c) {
            s += Z[(long)n * 1024 + b * 32 + c] * Wn[c * 32 + o];
            s += Y1[(long)n * 1024 + b * 32 + c] * Wn[1024 + c * 32 + o];
        }
        h[((long)b * 128 + o) * 2048 + n] = s;
    }
}

// o2 init: o2[b,c,t,n] = x[b,t,n,0]*cc_w[c] + cc_b[c]
__global__ void cc_k(const float* __restrict__ x, const float* __restrict__ ccw,
                     const float* __restrict__ ccb, float* __restrict__ o2) {
    long i = (long)blockIdx.x * 256 + threadIdx.x;  // b*PS + tp
    long b = i / PS, tp = i % PS;
    float v = x[(b * PS + tp) * 3];
    for (int c = 0; c < 32; ++c)
        o2[(b * 32 + c) * (long)PS + tp] = v * ccw[c] + ccb[c];
}

// conv1x1 over channels via WMMA. in:(B,IN,12,2048) -> out:(B,OUT,12,2048)
template <int IN_CH, int OUT_CH, bool GELU>
__global__ void conv1x1_k(const float* __restrict__ in, const float* __restrict__ w,
                          const float* __restrict__ bias, float* __restrict__ out) {
    constexpr int CHT = OUT_CH / 16;
    constexpr int PT = 8 / CHT;
    __shared__ float Wl[OUT_CH * IN_CH];
    __shared__ float Bl[OUT_CH];
    int tid = threadIdx.x;
    for (int i = tid; i < OUT_CH * IN_CH; i += 256) Wl[i] = w[i];
    for (int i = tid; i < OUT_CH; i += 256) Bl[i] = bias[i];
    __syncthreads();
    int wv = tid >> 5, ln = tid & 31;
    int l16 = ln & 15, half = ln >> 4, klo = 2 * half;
    int ct = wv % CHT, pt = wv / CHT;
    int b = blockIdx.y;
    long tp0 = (long)blockIdx.x * (PT * 16) + pt * 16;
    const float* inb = in + (long)b * IN_CH * PS;
    v8f acc = vzero8();
    for (int k = 0; k < IN_CH; k += 4) {
        v2f a, bm;
        a.x = Wl[(ct * 16 + l16) * IN_CH + k + klo];
        a.y = Wl[(ct * 16 + l16) * IN_CH + k + klo + 1];
        bm.x = inb[(long)(k + klo) * PS + tp0 + l16];
        bm.y = inb[(long)(k + klo + 1) * PS + tp0 + l16];
        acc = wmma4(a, bm, acc);
    }
    float* outb = out + (long)b * OUT_CH * PS;
    for (int r = 0; r < 8; ++r) {
        int m = r + 8 * half, oc = ct * 16 + m;
        float v = acc[r] + Bl[oc];
        if (GELU) v = gelu_exact(v);
        outb[(long)oc * PS + tp0 + l16] = v;
    }
}

// time-mixing conv (only i=1 filter pair is live in the reference):
// cm[b,h,tc,n] = tanh(F) * sigmoid(G) over 12 in-ch (time) x 9 taps (core dim)
__global__ void timeconv_k(const float* __restrict__ t2, const float* __restrict__ fw,
                           const float* __restrict__ fb, const float* __restrict__ gw,
                           const float* __restrict__ gb, float* __restrict__ cm) {
    __shared__ float Fl[1296], Gl[1296], Fb[12], Gb[12];
    int tid = threadIdx.x;
    for (int i = tid; i < 1296; i += 256) { Fl[i] = fw[i]; Gl[i] = gw[i]; }
    if (tid < 12) { Fb[tid] = fb[tid]; Gb[tid] = gb[tid]; }
    __syncthreads();
    int n = blockIdx.x * 256 + tid;
    int hh = blockIdx.y, b = blockIdx.z;
    float aF[12], aG[12];
    for (int o = 0; o < 12; ++o) { aF[o] = Fb[o]; aG[o] = Gb[o]; }
    for (int ii = 0; ii < 12; ++ii)
        for (int r = 0; r < 9; ++r) {
            float v = t2[(((long)b * 16 + hh + r) * 12 + ii) * 2048 + n];
            for (int o = 0; o < 12; ++o) {
                aF[o] += Fl[(o * 12 + ii) * 9 + r] * v;
                aG[o] += Gl[(o * 12 + ii) * 9 + r] * v;
            }
        }
    for (int o = 0; o < 12; ++o) {
        float f = tanhf(aF[o]);
        float g = 1.0f / (1.0f + expf(-aG[o]));
        cm[(((long)b * 8 + hh) * 12 + o) * 2048 + n] = f * g;
    }
}

// star tail: out = W4 @ gelu(W3 @ cat(inp[32], cm[8]) + b3) + b4
__global__ void star_tail_k(const float* __restrict__ inp, const float* __restrict__ cm,
                            const float* __restrict__ w3, const float* __restrict__ b3,
                            const float* __restrict__ w4, const float* __restrict__ b4,
                            float* __restrict__ out) {
    __shared__ float W3l[32 * 40], B3l[32], W4l[32 * 32], B4l[32];
    __shared__ float U[8][512];
    int tid = threadIdx.x;
    for (int i = tid; i < 32 * 40; i += 256) W3l[i] = w3[i];
    for (int i = tid; i < 32 * 32; i += 256) W4l[i] = w4[i];
    if (tid < 32) { B3l[tid] = b3[tid]; B4l[tid] = b4[tid]; }
    __syncthreads();
    int wv = tid >> 5, ln = tid & 31;
    int l16 = ln & 15, half = ln >> 4, klo = 2 * half;
    int b = blockIdx.y;
    long tp0 = (long)blockIdx.x * 128 + wv * 16;
    const float* inb = inp + (long)b * 32 * PS;
    const float* cmb = cm + (long)b * 8 * PS;
    for (int ct = 0; ct < 2; ++ct) {
        v8f acc = vzero8();
        for (int k = 0; k < 40; k += 4) {
            v2f a, bm;
            a.x = W3l[(ct * 16 + l16) * 40 + k + klo];
            a.y = W3l[(ct * 16 + l16) * 40 + k + klo + 1];
            int k0 = k + klo, k1 = k + klo + 1;
            bm.x = (k0 < 32) ? inb[(long)k0 * PS + tp0 + l16]
                             : cmb[(long)(k0 - 32) * PS + tp0 + l16];
            bm.y = (k1 < 32) ? inb[(long)k1 * PS + tp0 + l16]
                             : cmb[(long)(k1 - 32) * PS + tp0 + l16];
            acc = wmma4(a, bm, acc);
        }
        for (int r = 0; r < 8; ++r) {
            int m = r + 8 * half;
            U[wv][(ct * 16 + m) * 16 + l16] = gelu_exact(acc[r] + B3l[ct * 16 + m]);
        }
    }
    __syncthreads();
    float* outb = out + (long)b * 32 * PS;
    for (int ct = 0; ct < 2; ++ct) {
        v8f acc = vzero8();
        for (int k = 0; k < 32; k += 4) {
            v2f a, bm;
            a.x = W4l[(ct * 16 + l16) * 32 + k + klo];
            a.y = W4l[(ct * 16 + l16) * 32 + k + klo + 1];
            bm.x = U[wv][(k + klo) * 16 + l16];
            bm.y = U[wv][(k + klo + 1) * 16 + l16];
            acc = wmma4(a, bm, acc);
        }
        for (int r = 0; r < 8; ++r) {
            int m = r + 8 * half;
            outb[(long)(ct * 16 + m) * PS + tp0 + l16] = acc[r] + B4l[ct * 16 + m];
        }
    }
}

// h channels 32..127: star2 t=11 slice + embedding gathers
__global__ void assemble_h_k(const float* __restrict__ so, const float* __restrict__ x,
                             const float* __restrict__ te, const float* __restrict__ de,
                             float* __restrict__ h) {
    int i = blockIdx.x * 256 + threadIdx.x;  // b*2048+n
    int b = i >> 11, n = i & 2047;
    float* hb = h + (long)b * 128 * 2048 + n;
    const float* sp = so + (long)b * 32 * PS + 11 * 2048 + n;
    for (int c = 0; c < 32; ++c) hb[(32 + c) * 2048] = sp[(long)c * PS];
    const float* xr = x + (((long)b * 12 + 11) * 2048 + n) * 3;
    int ti = (int)(xr[1] * 288.0f); ti = ti < 0 ? 0 : (ti > 287 ? 287 : ti);
    int di = (int)(xr[2] * 7.0f);   di = di < 0 ? 0 : (di > 6 ? 6 : di);
    for (int c = 0; c < 32; ++c) hb[(64 + c) * 2048] = te[ti * 32 + c];
    for (int c = 0; c < 32; ++c) hb[(96 + c) * 2048] = de[di * 32 + c];
}

// one encoder layer, in place: h += W2 @ relu(W1 @ h + b1) + b2  (128x128 per column)
__global__ void enc_layer_k(float* __restrict__ h, const float* __restrict__ w1,
                            const float* __restrict__ b1, const float* __restrict__ w2,
                            const float* __restrict__ b2) {
    __shared__ float U[4][2048];  // 32KB: per-wave 128x16 activations
    int tid = threadIdx.x, wv = tid >> 5, ln = tid & 31;
    int l16 = ln & 15, half = ln >> 4, klo = 2 * half;
    int g = blockIdx.x * 4 + wv;
    int b = g >> 7, nt = g & 127;
    long n0 = (long)nt * 16;
    float* hb = h + (long)b * 128 * 2048;
    for (int mt = 0; mt < 8; ++mt) {
        v8f acc = vzero8();
        for (int k = 0; k < 128; k += 4) {
            v2f a, bm;
            a.x = w1[(mt * 16 + l16) * 128 + k + klo];
            a.y = w1[(mt * 16 + l16) * 128 + k + klo + 1];
            bm.x = hb[(long)(k + klo) * 2048 + n0 + l16];
            bm.y = hb[(long)(k + klo + 1) * 2048 + n0 + l16];
            acc = wmma4(a, bm, acc);
        }
        for (int r = 0; r < 8; ++r) {
            int m = r + 8 * half;
            float v = acc[r] + b1[mt * 16 + m];
            U[wv][(mt * 16 + m) * 16 + l16] = v > 0.0f ? v : 0.0f;
        }
    }
    __syncthreads();
    for (int mt = 0; mt < 8; ++mt) {
        v8f acc = vzero8();
        for (int k = 0; k < 128; k += 4) {
            v2f a, bm;
            a.x = w2[(mt * 16 + l16) * 128 + k + klo];
            a.y = w2[(mt * 16 + l16) * 128 + k + klo + 1];
            bm.x = U[wv][(k + klo) * 16 + l16];
            bm.y = U[wv][(k + klo + 1) * 16 + l16];
            acc = wmma4(a, bm, acc);
        }
        for (int r = 0; r < 8; ++r) {
            int m = r + 8 * half;
            long idx = (long)(mt * 16 + m) * 2048 + n0 + l16;
            hb[idx] = hb[idx] + acc[r] + b2[mt * 16 + m];
        }
    }
}

// final regression: out[b,p,n] = reg_w[p,:] . h[b,:,n] + reg_b[p]
__global__ void reg_k(const float* __restrict__ h, const float* __restrict__ rw,
                      const float* __restrict__ rb, float* __restrict__ out) {
    int i = blockIdx.x * 256 + threadIdx.x;  // b*2048+n
    int b = i >> 11, n = i & 2047;
    const float* hb = h + (long)b * 128 * 2048 + n;
    float acc[12];
    for (int p = 0; p < 12; ++p) acc[p] = rb[p];
    for (int ch = 0; ch < 128; ++ch) {
        float v = hb[(long)ch * 2048];
        for (int p = 0; p < 12; ++p) acc[p] += rw[p * 128 + ch] * v;
    }
    for (int p = 0; p < 12; ++p) out[(((long)b * 12 + p) * 2048) + n] = acc[p];
}

// ---------------------------------------------------------------------------
extern "C" void kernel_launch(void* const* d_in, const int* in_sizes, int n_in,
                              void* d_out, int out_size, void* d_ws, size_t ws_size,
                              hipStream_t stream) {
    (void)in_sizes; (void)n_in; (void)out_size; (void)ws_size;
    const float* x        = (const float*)d_in[0];
    const float* node_emb = (const float*)d_in[1];
    const float* tid_emb  = (const float*)d_in[2];
    const float* diw_emb  = (const float*)d_in[3];
    const float* gcn_w0   = (const float*)d_in[4];
    const float* gcn_b0   = (const float*)d_in[5];
    const float* gcn_w1   = (const float*)d_in[6];
    const float* gcn_b1   = (const float*)d_in[7];
    const float* cc_w     = (const float*)d_in[8];
    const float* cc_b     = (const float*)d_in[9];
    const float* s_w1     = (const float*)d_in[10];
    const float* s_b1     = (const float*)d_in[11];
    const float* s_w2     = (const float*)d_in[12];
    const float* s_b2     = (const float*)d_in[13];
    const float* s_w3     = (const float*)d_in[14];
    const float* s_b3     = (const float*)d_in[15];
    const float* s_w4     = (const float*)d_in[16];
    const float* s_b4     = (const float*)d_in[17];
    const float* s_fw     = (const float*)d_in[18];
    const float* s_fb     = (const float*)d_in[19];
    const float* s_gw     = (const float*)d_in[20];
    const float* s_gb     = (const float*)d_in[21];
    const float* e_w1     = (const float*)d_in[22];
    const float* e_b1     = (const float*)d_in[23];
    const float* e_w2     = (const float*)d_in[24];
    const float* e_b2     = (const float*)d_in[25];
    const float* reg_w    = (const float*)d_in[26];
    const float* reg_b    = (const float*)d_in[27];

    float* ws = (float*)d_ws;
    float* sbuf0 = ws;                 // 25165824 floats (B,32,12,2048)
    float* sbuf1 = ws + 25165824L;     // 25165824
    float* t2b   = ws + 50331648L;     // 12582912 (B,16,12,2048)
    float* cmb   = ws + 62914560L;     //  6291456 (B,8,12,2048)
    float* hbuf  = ws + 69206016L;     //  8388608 (B,128,2048)
    // GCN scratch overlaid on sbuf0 region (GCN finishes before stars start)
    float* Adj = ws;                   // 4194304
    float* X0  = ws + 4194304L;        // 65536
    float* Y0  = ws + 4259840L;        // 65536
    float* Zb  = ws + 4325376L;        // 2097152
    float* Y1  = ws + 6422528L;        // 2097152

    // --- GCN branch (only t=11 survives the reference's final slice) ---
    adj_k<<<2048, 256, 0, stream>>>(node_emb, Adj);
    pack_x0_k<<<256, 256, 0, stream>>>(x, X0);
    gemm_k<<<64, 128, 0, stream>>>(Adj, X0, Y0, 2048, 32, 2);          // y0 = A @ X0
    gcn0_k<<<2048, 32, 0, stream>>>(node_emb, gcn_w0, gcn_b0, X0, Y0, Zb);
    gemm_k<<<2048, 128, 0, stream>>>(Adj, Zb, Y1, 2048, 1024, 64);     // y1 = A @ Z
    gcn1_k<<<2048, 256, 0, stream>>>(node_emb, gcn_w1, gcn_b1, Zb, Y1, hbuf);

    // --- STAR branch ---
    cc_k<<<3072, 256, 0, stream>>>(x, cc_w, cc_b, sbuf0);
    // star 0: sbuf0 -> sbuf1
    conv1x1_k<32, 32, true ><<<dim3(384, 32), 256, 0, stream>>>(sbuf0, s_w1, s_b1, sbuf1);
    conv1x1_k<32, 16, false><<<dim3(192, 32), 256, 0, stream>>>(sbuf1, s_w2, s_b2, t2b);
    timeconv_k<<<dim3(8, 8, 32), 256, 0, stream>>>(t2b, s_fw + 1296, s_fb + 12,
                                                   s_gw + 1296, s_gb + 12, cmb);
    star_tail_k<<<dim3(192, 32), 256, 0, stream>>>(sbuf0, cmb, s_w3, s_b3, s_w4, s_b4, sbuf1);
    // star 1: sbuf1 -> sbuf0
    conv1x1_k<32, 32, true ><<<dim3(384, 32), 256, 0, stream>>>(sbuf1, s_w1 + 1024, s_b1 + 32, sbuf0);
    conv1x1_k<32, 16, false><<<dim3(192, 32), 256, 0, stream>>>(sbuf0, s_w2 + 512, s_b2 + 16, t2b);
    timeconv_k<<<dim3(8, 8, 32), 256, 0, stream>>>(t2b, s_fw + 3 * 1296, s_fb + 3 * 12,
                                                   s_gw + 3 * 1296, s_gb + 3 * 12, cmb);
    star_tail_k<<<dim3(192, 32), 256, 0, stream>>>(sbuf1, cmb, s_w3 + 1280, s_b3 + 32,
                                                   s_w4 + 1024, s_b4 + 32, sbuf0);

    // --- fuse + encoder + regression ---
    assemble_h_k<<<256, 256, 0, stream>>>(sbuf0, x, tid_emb, diw_emb, hbuf);
    for (int l = 0; l < 3; ++l)
        enc_layer_k<<<1024, 128, 0, stream>>>(hbuf, e_w1 + (long)l * 16384, e_b1 + l * 128,
                                              e_w2 + (long)l * 16384, e_b2 + l * 128);
    reg_k<<<256, 256, 0, stream>>>(hbuf, reg_w, reg_b, (float*)d_out);
}